// SupConLoss_60885456388289
// MI455X (gfx1250) — compile-verified
//
#include <hip/hip_runtime.h>
#include <math.h>

// ---------------- problem constants ----------------
#define BSZ   1024
#define N2    2048      // n = bsz * n_views
#define DIM   192
#define NC    2793
#define EPSF  1e-12f

// ArcFace constants (m = 0.2)
#define COS_M  0.9800665778412416f
#define SIN_M  0.19866933079506122f
#define TH_C   (-0.9800665778412416f)
#define MM_C   0.03973386615901224f
#define S_C    10.0f
#define TEMP_C 0.07f

typedef __attribute__((ext_vector_type(2))) float v2f;
typedef __attribute__((ext_vector_type(8))) float v8f;

__device__ __forceinline__ v2f ld2(const float* p) {
    return *reinterpret_cast<const v2f*>(p);
}

#define WMMA_F32(a, b, c) \
    __builtin_amdgcn_wmma_f32_16x16x4_f32(false, (a), false, (b), (short)0, (c), false, false)

// ---------------- kernel 0: zero the scalar output ----------------
__global__ void k_zero(float* out) {
    if (threadIdx.x == 0 && blockIdx.x == 0) out[0] = 0.0f;
}

// ---------------- kernel 1: build cf (stack views) + L2 normalize rows ----------------
// features layout (bsz, n_views=2, d): cf[i] = features[i&1023, i>>10, :]
__global__ __launch_bounds__(256) void k_norm_cf(const float* __restrict__ feats,
                                                 float* __restrict__ cfn) {
    const int i = blockIdx.x;
    const int t = threadIdx.x;
    const float* src = feats + ((size_t)((i & (BSZ - 1)) * 2 + (i >> 10))) * DIM;
    float v = (t < DIM) ? src[t] : 0.0f;
    __shared__ float red[256];
    red[t] = v * v;
    __syncthreads();
    for (int w = 128; w > 0; w >>= 1) {
        if (t < w) red[t] += red[t + w];
        __syncthreads();
    }
    const float scale = 1.0f / fmaxf(sqrtf(red[0]), EPSF);
    if (t < DIM) cfn[(size_t)i * DIM + t] = v * scale;
}

// ---------------- shared GEMM core: 32x32 wave tile, f32 WMMA 16x16x4 ----------------
// 4 accumulators; each k-step: 4 fragment loads feed 4 WMMAs (each fragment used twice).
__device__ __forceinline__ void gemm32x32_f32(const float* __restrict__ arow0,
                                              const float* __restrict__ arow1,
                                              const float* __restrict__ brow0,
                                              const float* __restrict__ brow1,
                                              int half,
                                              v8f& acc00, v8f& acc01,
                                              v8f& acc10, v8f& acc11) {
    for (int k = 0; k < DIM; k += 4) {
        const int kk = k + half * 2;      // lane half selects K pair (K0,K1 | K2,K3)
        v2f a0 = ld2(arow0 + kk);
        v2f a1 = ld2(arow1 + kk);
        v2f b0 = ld2(brow0 + kk);
        v2f b1 = ld2(brow1 + kk);
        acc00 = WMMA_F32(a0, b0, acc00);
        acc01 = WMMA_F32(a0, b1, acc01);
        acc10 = WMMA_F32(a1, b0, acc10);
        acc11 = WMMA_F32(a1, b1, acc11);
    }
}

// ---------------- kernel 2: cosine = cfn @ cfn^T ----------------
// block: 8 waves as 2(M) x 4(N) of 32x32 wave tiles -> 64 x 128; grid (16, 32)
__global__ __launch_bounds__(256) void k_gemm_cos(const float* __restrict__ cfn,
                                                  float* __restrict__ cosm) {
    const int lane = threadIdx.x & 31;
    const int wave = threadIdx.x >> 5;
    const int half = lane >> 4;
    const int r    = lane & 15;
    const int m0 = blockIdx.y * 64  + (wave >> 2) * 32;
    const int n0 = blockIdx.x * 128 + (wave & 3) * 32;

    const float* arow0 = cfn + (size_t)(m0 + r) * DIM;
    const float* arow1 = cfn + (size_t)(m0 + 16 + r) * DIM;
    const float* brow0 = cfn + (size_t)(n0 + r) * DIM;          // B = cfn^T -> rows of cfn
    const float* brow1 = cfn + (size_t)(n0 + 16 + r) * DIM;

    v8f acc00 = {}, acc01 = {}, acc10 = {}, acc11 = {};
    gemm32x32_f32(arow0, arow1, brow0, brow1, half, acc00, acc01, acc10, acc11);

    // C/D layout: VGPR rr -> M = rr + 8*half ; N = lane&15
    for (int rr = 0; rr < 8; ++rr) {
        const int row0 = m0 + rr + half * 8;
        const int row1 = row0 + 16;
        const int col  = n0 + r;
        cosm[(size_t)row0 * N2 + col]      = acc00[rr];
        cosm[(size_t)row0 * N2 + col + 16] = acc01[rr];
        cosm[(size_t)row1 * N2 + col]      = acc10[rr];
        cosm[(size_t)row1 * N2 + col + 16] = acc11[rr];
    }
}

// ---------------- kernel 3: h2 = tanh(cfn @ fc_w[:2048]^T + fc_b) ----------------
__global__ __launch_bounds__(256) void k_gemm_h2(const float* __restrict__ cfn,
                                                 const float* __restrict__ fcw,
                                                 const float* __restrict__ fcb,
                                                 float* __restrict__ h2) {
    const int lane = threadIdx.x & 31;
    const int wave = threadIdx.x >> 5;
    const int half = lane >> 4;
    const int r    = lane & 15;
    const int m0 = blockIdx.y * 64  + (wave >> 2) * 32;
    const int n0 = blockIdx.x * 128 + (wave & 3) * 32;

    const float* arow0 = cfn + (size_t)(m0 + r) * DIM;
    const float* arow1 = cfn + (size_t)(m0 + 16 + r) * DIM;
    const float* brow0 = fcw + (size_t)(n0 + r) * DIM;          // B = fc_w^T -> rows of fc_w
    const float* brow1 = fcw + (size_t)(n0 + 16 + r) * DIM;

    v8f acc00 = {}, acc01 = {}, acc10 = {}, acc11 = {};
    gemm32x32_f32(arow0, arow1, brow0, brow1, half, acc00, acc01, acc10, acc11);

    for (int rr = 0; rr < 8; ++rr) {
        const int row0 = m0 + rr + half * 8;
        const int row1 = row0 + 16;
        const int col  = n0 + r;
        h2[(size_t)row0 * N2 + col]      = tanhf(acc00[rr] + fcb[col]);
        h2[(size_t)row0 * N2 + col + 16] = tanhf(acc01[rr] + fcb[col + 16]);
        h2[(size_t)row1 * N2 + col]      = tanhf(acc10[rr] + fcb[col]);
        h2[(size_t)row1 * N2 + col + 16] = tanhf(acc11[rr] + fcb[col + 16]);
    }
}

// ---------------- kernel 4: row L2 norms of cosine ----------------
__global__ __launch_bounds__(256) void k_rownorm(const float* __restrict__ cosm,
                                                 float* __restrict__ rn) {
    const int i = blockIdx.x;
    const int t = threadIdx.x;
    const float* row = cosm + (size_t)i * N2;
    float s = 0.0f;
    for (int k = t; k < N2; k += 256) {
        const float v = row[k];
        s += v * v;
    }
    __shared__ float red[256];
    red[t] = s;
    __syncthreads();
    for (int w = 128; w > 0; w >>= 1) {
        if (t < w) red[t] += red[t + w];
        __syncthreads();
    }
    if (t == 0) rn[i] = sqrtf(red[0]);
}

// ---------------- kernel 5: group indices by label (deterministic) ----------------
__global__ __launch_bounds__(256) void k_build_order(const int* __restrict__ labels,
                                                     int* counts, int* offs, int* order,
                                                     int* present, int* pcount) {
    const int t = threadIdx.x;
    for (int c = t; c < NC; c += 256) counts[c] = 0;
    __syncthreads();
    for (int i = t; i < N2; i += 256) atomicAdd(&counts[labels[i & (BSZ - 1)]], 1);
    __syncthreads();
    if (t == 0) {
        int run = 0, P = 0;
        for (int c = 0; c < NC; ++c) {
            offs[c] = run;
            run += counts[c];
            if (counts[c] > 0) present[P++] = c;
        }
        *pcount = P;
        for (int i = 0; i < N2; ++i) {        // serial, deterministic scatter
            const int c = labels[i & (BSZ - 1)];
            order[offs[c]++] = i;
        }
    }
}

// ---------------- kernel 6: atten[j] from per-class sums of h2 row j ----------------
__global__ __launch_bounds__(256) void k_atten(const float* __restrict__ h2,
                                               const int* __restrict__ labels,
                                               const int* __restrict__ counts,
                                               const int* __restrict__ offs,
                                               const int* __restrict__ order,
                                               const int* __restrict__ present,
                                               const int* __restrict__ pcount,
                                               float* __restrict__ atten) {
    const int j = blockIdx.x;
    const int t = threadIdx.x;
    const int P = *pcount;
    const int myc = labels[j & (BSZ - 1)];
    const float* hrow = h2 + (size_t)j * N2;
    __shared__ float s_num;
    __shared__ float red[256];
    if (t == 0) s_num = 0.0f;
    __syncthreads();
    float sumExp = 0.0f;
    for (int p = t; p < P; p += 256) {
        const int c = present[p];
        const int cnt = counts[c];
        const int start = offs[c] - cnt;
        float s = 0.0f;
        for (int m = 0; m < cnt; ++m) s += hrow[order[start + m]];
        const float e = expf(s);
        sumExp += e;
        if (c == myc) s_num = e;   // exactly one thread hits this
    }
    red[t] = sumExp;
    __syncthreads();
    for (int w = 128; w > 0; w >>= 1) {
        if (t < w) red[t] += red[t + w];
        __syncthreads();
    }
    if (t == 0) {
        const float denom = (float)(NC - P) + red[0];   // empty classes contribute exp(0)=1
        const float a = s_num / denom;
        atten[j] = a / fmaxf(fabsf(a), EPSF);
    }
}

// ---------------- kernel 7: ArcFace phi + scale, in-place; row max ----------------
__global__ __launch_bounds__(256) void k_phi_rowmax(float* __restrict__ cosm,
                                                    const float* __restrict__ rn,
                                                    float* __restrict__ rowmax) {
    const int i = blockIdx.x;
    const int t = threadIdx.x;
    const float scale = 1.0f / fmaxf(rn[i], EPSF);
    float* row = cosm + (size_t)i * N2;
    float mx = -INFINITY;
    for (int k = t; k < N2; k += 256) {
        const float cn = row[k] * scale;
        float s2 = 1.0f - cn * cn;
        s2 = fminf(fmaxf(s2, 0.0f), 1.0f);
        const float sine = sqrtf(s2);
        float phi = cn * COS_M - sine * SIN_M;
        phi = (cn - TH_C > 0.0f) ? phi : (cn - MM_C);
        const float adc = (S_C * phi) / TEMP_C;   // anchor_dot_contrast
        row[k] = adc;
        mx = fmaxf(mx, adc);
    }
    __shared__ float red[256];
    red[t] = mx;
    __syncthreads();
    for (int w = 128; w > 0; w >>= 1) {
        if (t < w) red[t] = fmaxf(red[t], red[t + w]);
        __syncthreads();
    }
    if (t == 0) rowmax[i] = red[0];
}

// ---------------- kernel 8: per-row masked LSE + loss accumulation ----------------
__global__ __launch_bounds__(256) void k_loss(const float* __restrict__ cosm,
                                              const float* __restrict__ rowmax,
                                              const float* __restrict__ atten,
                                              const int* __restrict__ labels,
                                              float* __restrict__ out) {
    const int i = blockIdx.x;
    const int t = threadIdx.x;
    const float att = atten[i];
    const float m = rowmax[i];
    const int li = labels[i & (BSZ - 1)];
    const float* row = cosm + (size_t)i * N2;
    float esum = 0.0f, psum = 0.0f;
    int cnt = 0;
    for (int k = t; k < N2; k += 256) {
        const float l = (row[k] - m) * att;
        if (k != i) {
            esum += expf(l);                        // logits_mask excludes diagonal
            if (labels[k & (BSZ - 1)] == li) {      // mask_full positives
                psum += l;
                cnt += 1;
            }
        }
    }
    __shared__ float rE[256];
    __shared__ float rP[256];
    __shared__ int rC[256];
    rE[t] = esum; rP[t] = psum; rC[t] = cnt;
    __syncthreads();
    for (int w = 128; w > 0; w >>= 1) {
        if (t < w) { rE[t] += rE[t + w]; rP[t] += rP[t + w]; rC[t] += rC[t + w]; }
        __syncthreads();
    }
    if (t == 0) {
        const float logZ = logf(rE[0]);
        const float mlpp = rP[0] / (float)rC[0] - logZ;
        const float lossi = -mlpp;                   // TEMP/BASE_TEMP == 1
        atomicAdd(out, lossi * (1.0f / (float)N2));
    }
}

// ---------------- launch ----------------
extern "C" void kernel_launch(void* const* d_in, const int* in_sizes, int n_in,
                              void* d_out, int out_size, void* d_ws, size_t ws_size,
                              hipStream_t stream) {
    const float* feats  = (const float*)d_in[0];   // (1024, 2, 192)
    const int*   labels = (const int*)  d_in[1];   // (1024,)
    const float* fcw    = (const float*)d_in[2];   // (2793, 192)
    const float* fcb    = (const float*)d_in[3];   // (2793,)
    float* out = (float*)d_out;

    // workspace layout (~34.4 MB)
    float* cfn  = (float*)d_ws;                    // 2048*192
    float* cosm = cfn  + (size_t)N2 * DIM;         // 2048*2048 (later holds adc)
    float* h2   = cosm + (size_t)N2 * N2;          // 2048*2048
    float* rn   = h2   + (size_t)N2 * N2;          // 2048
    float* rmx  = rn   + N2;                       // 2048
    float* att  = rmx  + N2;                       // 2048
    int* counts  = (int*)(att + N2);               // 2793
    int* offs    = counts  + NC;                   // 2793
    int* order   = offs    + NC;                   // 2048
    int* present = order   + N2;                   // <=2793
    int* pcount  = present + NC;                   // 1

    k_zero<<<1, 64, 0, stream>>>(out);
    k_norm_cf<<<N2, 256, 0, stream>>>(feats, cfn);

    dim3 gg(N2 / 128, N2 / 64);                    // 16 x 32 blocks, 256 threads
    k_gemm_cos<<<gg, 256, 0, stream>>>(cfn, cosm);
    k_gemm_h2<<<gg, 256, 0, stream>>>(cfn, fcw, fcb, h2);

    k_rownorm<<<N2, 256, 0, stream>>>(cosm, rn);
    k_build_order<<<1, 256, 0, stream>>>(labels, counts, offs, order, present, pcount);
    k_atten<<<N2, 256, 0, stream>>>(h2, labels, counts, offs, order, present, pcount, att);
    k_phi_rowmax<<<N2, 256, 0, stream>>>(cosm, rn, rmx);
    k_loss<<<N2, 256, 0, stream>>>(cosm, rmx, att, labels, out);
}